// MoMuGNN_25675314495589
// MI455X (gfx1250) — compile-verified
//
#include <hip/hip_runtime.h>

typedef __attribute__((ext_vector_type(2))) float v2f;
typedef __attribute__((ext_vector_type(8))) float v8f;

#define WMMA_F32X4(a, b, c) \
  __builtin_amdgcn_wmma_f32_16x16x4_f32(false, (a), false, (b), (short)0, (c), false, false)

#define TM 32           // rows per block tile
#define LDW1 132        // padded LDS stride for 128-wide input tile
#define LDW2 260        // padded LDS stride for 256-wide mid tile
#define NEG_HUGE (-3.4028234663852886e38f)

// ---------------------------------------------------------------- utilities
__global__ void zero_f32(float* __restrict__ p, long long n4) {
  long long i = (long long)blockIdx.x * blockDim.x + threadIdx.x;
  if (i < n4) ((float4*)p)[i] = make_float4(0.f, 0.f, 0.f, 0.f);
}

// -------------------------------------------------- edge scatter-add (GIN aggr)
// thread t = (edge e, feature-quad q); agg[dst] += h[src]
__global__ void scatter_add_k(const float* __restrict__ h, float* __restrict__ agg,
                              const long long* __restrict__ ei, long long E) {
  long long t = (long long)blockIdx.x * blockDim.x + threadIdx.x;
  if (t >= E * 32) return;
  long long e = t >> 5;
  int q = (int)(t & 31);
  long long s = ei[e];
  long long d = ei[E + e];
  float4 v = ((const float4*)(h + s * 128))[q];
  float* b = agg + d * 128 + (long long)q * 4;
  atomicAdd(b + 0, v.x);
  atomicAdd(b + 1, v.y);
  atomicAdd(b + 2, v.z);
  atomicAdd(b + 3, v.w);
}

// -------------------------------------------------- fused MLP (two fp32 WMMA GEMMs)
// z = relu((h+agg) @ W1 + b1) @ W2 + b2 ; also accumulate per-feature sum/sumsq
__global__ __launch_bounds__(256) void mlp_k(
    const float* __restrict__ h, const float* __restrict__ agg,
    const float* __restrict__ W1, const float* __restrict__ b1,
    const float* __restrict__ W2, const float* __restrict__ b2,
    float* __restrict__ z, float* __restrict__ sums, float* __restrict__ sumsq,
    int N) {
  __shared__ float lz[TM * LDW1];   // input tile  [32 x 128] (padded)
  __shared__ float lm[TM * LDW2];   // mid tile    [32 x 256] (padded)
  __shared__ float cs[128];
  __shared__ float cs2[128];

  const int tid = threadIdx.x;
  const int wave = tid >> 5;
  const int lane = tid & 31;
  const int row0 = blockIdx.x * TM;

  // load input tile (h + agg), zero-fill rows >= N
  for (int i = tid; i < TM * 32; i += 256) {
    int r = i >> 5, q = i & 31;
    int gr = row0 + r;
    float4 v = make_float4(0.f, 0.f, 0.f, 0.f);
    if (gr < N) {
      float4 a = ((const float4*)(h + (long long)gr * 128))[q];
      float4 g = ((const float4*)(agg + (long long)gr * 128))[q];
      v.x = a.x + g.x; v.y = a.y + g.y; v.z = a.z + g.z; v.w = a.w + g.w;
    }
    float* d = &lz[r * LDW1 + q * 4];
    d[0] = v.x; d[1] = v.y; d[2] = v.z; d[3] = v.w;
  }
  if (tid < 128) { cs[tid] = 0.f; cs2[tid] = 0.f; }
  __syncthreads();

  const int mlane = lane & 15;          // row (A) / col (B,C/D) within 16
  const int kk = (lane >> 4) * 2;       // K sub-offset per ISA 32-bit layouts
  const int rbase = (lane >> 4) * 8;    // C/D row group

  // ---- GEMM1: [32x128] @ [128x256] -> lm ; wave handles N-tiles {wave, wave+8}
  for (int nt = 0; nt < 2; ++nt) {
    int n0 = (wave + nt * 8) * 16;
    v8f acc0 = {}; v8f acc1 = {};
    for (int k0 = 0; k0 < 128; k0 += 4) {
      v2f bf;
      bf.x = W1[(k0 + kk) * 256 + n0 + mlane];
      bf.y = W1[(k0 + kk + 1) * 256 + n0 + mlane];
      v2f a0, a1;
      a0.x = lz[mlane * LDW1 + k0 + kk];
      a0.y = lz[mlane * LDW1 + k0 + kk + 1];
      a1.x = lz[(16 + mlane) * LDW1 + k0 + kk];
      a1.y = lz[(16 + mlane) * LDW1 + k0 + kk + 1];
      acc0 = WMMA_F32X4(a0, bf, acc0);
      acc1 = WMMA_F32X4(a1, bf, acc1);
    }
    float bias = b1[n0 + mlane];
    for (int r = 0; r < 8; ++r) {
      float v0 = acc0[r] + bias; v0 = v0 > 0.f ? v0 : 0.f;
      lm[(rbase + r) * LDW2 + n0 + mlane] = v0;
      float v1 = acc1[r] + bias; v1 = v1 > 0.f ? v1 : 0.f;
      lm[(16 + rbase + r) * LDW2 + n0 + mlane] = v1;
    }
  }
  __syncthreads();

  // ---- GEMM2: [32x256] @ [256x128] -> z ; wave handles N-tile = wave
  {
    int n0 = wave * 16;
    v8f acc0 = {}; v8f acc1 = {};
    for (int k0 = 0; k0 < 256; k0 += 4) {
      v2f bf;
      bf.x = W2[(k0 + kk) * 128 + n0 + mlane];
      bf.y = W2[(k0 + kk + 1) * 128 + n0 + mlane];
      v2f a0, a1;
      a0.x = lm[mlane * LDW2 + k0 + kk];
      a0.y = lm[mlane * LDW2 + k0 + kk + 1];
      a1.x = lm[(16 + mlane) * LDW2 + k0 + kk];
      a1.y = lm[(16 + mlane) * LDW2 + k0 + kk + 1];
      acc0 = WMMA_F32X4(a0, bf, acc0);
      acc1 = WMMA_F32X4(a1, bf, acc1);
    }
    float bias = b2[n0 + mlane];
    float s = 0.f, s2 = 0.f;
    for (int r = 0; r < 8; ++r) {
      int gr0 = row0 + rbase + r;
      float v0 = acc0[r] + bias;
      if (gr0 < N) {
        z[(long long)gr0 * 128 + n0 + mlane] = v0;
        s += v0; s2 += v0 * v0;
      }
      int gr1 = row0 + 16 + rbase + r;
      float v1 = acc1[r] + bias;
      if (gr1 < N) {
        z[(long long)gr1 * 128 + n0 + mlane] = v1;
        s += v1; s2 += v1 * v1;
      }
    }
    // lanes l and l+16 hold the same output column -> combine, then per-wave
    // columns are unique, so plain LDS accumulate is race-free.
    s  += __shfl_down(s, 16, 32);
    s2 += __shfl_down(s2, 16, 32);
    if (lane < 16) { cs[n0 + mlane] += s; cs2[n0 + mlane] += s2; }
  }
  __syncthreads();
  if (tid < 128) {
    atomicAdd(&sums[tid], cs[tid]);
    atomicAdd(&sumsq[tid], cs2[tid]);
  }
}

// -------------------------------------------------- BatchNorm finalize + apply
__global__ void bn_finalize_k(const float* __restrict__ sums, const float* __restrict__ sumsq,
                              const float* __restrict__ gamma, const float* __restrict__ beta,
                              float* __restrict__ scale, float* __restrict__ shift, int N) {
  int f = threadIdx.x;
  float invN = 1.0f / (float)N;
  float mean = sums[f] * invN;
  float var = sumsq[f] * invN - mean * mean;
  float inv = rsqrtf(var + 1e-5f);
  float sc = gamma[f] * inv;
  scale[f] = sc;
  shift[f] = beta[f] - mean * sc;
}

__global__ void bn_apply_k(float* __restrict__ z, const float* __restrict__ scale,
                           const float* __restrict__ shift, long long n4, int relu) {
  long long i = (long long)blockIdx.x * blockDim.x + threadIdx.x;
  if (i >= n4) return;
  int f = (int)(i & 31) * 4;
  float4 v = ((float4*)z)[i];
  v.x = v.x * scale[f + 0] + shift[f + 0];
  v.y = v.y * scale[f + 1] + shift[f + 1];
  v.z = v.z * scale[f + 2] + shift[f + 2];
  v.w = v.w * scale[f + 3] + shift[f + 3];
  if (relu) {
    v.x = v.x > 0.f ? v.x : 0.f;
    v.y = v.y > 0.f ? v.y : 0.f;
    v.z = v.z > 0.f ? v.z : 0.f;
    v.w = v.w > 0.f ? v.w : 0.f;
  }
  ((float4*)z)[i] = v;
}

// -------------------------------------------------- graph ranges (batch is sorted)
__global__ void graph_ranges_k(const long long* __restrict__ batch, int* __restrict__ start,
                               int N, int G) {
  int i = blockIdx.x * blockDim.x + threadIdx.x;
  if (i >= N) return;
  int g = (int)batch[i];
  if (i == 0) {
    for (int k = 0; k <= g; ++k) start[k] = 0;
  } else {
    int gp = (int)batch[i - 1];
    for (int k = gp + 1; k <= g; ++k) start[k] = i;
  }
  if (i == N - 1) {
    for (int k = g + 1; k <= G; ++k) start[k] = N;
  }
}

// -------------------------------------------------- global max pool: block=graph
__global__ void pool_max_k(const float* __restrict__ h, const int* __restrict__ start,
                           float* __restrict__ out) {
  int g = blockIdx.x;
  int f = threadIdx.x;  // 128 threads
  int s = start[g], e = start[g + 1];
  float m = NEG_HUGE;
  for (int i = s; i < e; ++i) m = fmaxf(m, h[(long long)i * 128 + f]);
  out[(long long)g * 128 + f] = m;
}

// ================================================================ host side
extern "C" void kernel_launch(void* const* d_in, const int* in_sizes, int n_in,
                              void* d_out, int out_size, void* d_ws, size_t ws_size,
                              hipStream_t stream) {
  const float* x = (const float*)d_in[0];
  const long long* ei = (const long long*)d_in[1];
  const long long* batch = (const long long*)d_in[2];
  const float* W1 = (const float*)d_in[4];
  const float* b1 = (const float*)d_in[5];
  const float* W2 = (const float*)d_in[6];
  const float* b2 = (const float*)d_in[7];
  const float* gamma = (const float*)d_in[8];
  const float* beta = (const float*)d_in[9];
  float* out = (float*)d_out;

  const int F = 128, H = 256;
  const int N = in_sizes[0] / F;
  const long long E = (long long)in_sizes[1] / 2;
  const int G = out_size / F;
  const int L = in_sizes[4] / (F * H);

  // workspace carve-up
  float* bufA = (float*)d_ws;
  float* bufB = bufA + (long long)N * F;
  float* agg  = bufB + (long long)N * F;
  float* sums  = agg + (long long)N * F;   // 128
  float* sumsq = sums + 128;               // 128 (contiguous with sums)
  float* scale = sumsq + 128;              // 128
  float* shift = scale + 128;              // 128
  int* gstart  = (int*)(shift + 128);      // G+1

  const long long nodes4 = (long long)N * (F / 4);
  dim3 blk(256);
  dim3 gridZeroAgg((unsigned)((nodes4 + 255) / 256));
  dim3 gridScatter((unsigned)((E * 32 + 255) / 256));
  dim3 gridMlp((unsigned)((N + TM - 1) / TM));
  dim3 gridBn((unsigned)((nodes4 + 255) / 256));

  const float* cur = x;
  for (int l = 0; l < L; ++l) {
    float* zbuf = (l & 1) ? bufB : bufA;
    // zero agg and stats
    zero_f32<<<gridZeroAgg, blk, 0, stream>>>(agg, nodes4);
    zero_f32<<<dim3(1), dim3(64), 0, stream>>>(sums, 64);  // 256 floats = 64 float4
    // aggregation
    scatter_add_k<<<gridScatter, blk, 0, stream>>>(cur, agg, ei, E);
    // fused MLP (WMMA) + BN stats
    mlp_k<<<gridMlp, blk, 0, stream>>>(cur, agg,
                                       W1 + (long long)l * F * H, b1 + (long long)l * H,
                                       W2 + (long long)l * H * F, b2 + (long long)l * F,
                                       zbuf, sums, sumsq, N);
    // batchnorm
    bn_finalize_k<<<dim3(1), dim3(128), 0, stream>>>(sums, sumsq, gamma + (long long)l * F,
                                                     beta + (long long)l * F, scale, shift, N);
    bn_apply_k<<<gridBn, blk, 0, stream>>>(zbuf, scale, shift, nodes4, (l < L - 1) ? 1 : 0);
    cur = zbuf;
  }

  // global max pool
  graph_ranges_k<<<dim3((N + 255) / 256), blk, 0, stream>>>(batch, gstart, N, G);
  pool_max_k<<<dim3(G), dim3(128), 0, stream>>>(cur, gstart, out);
}